// HypergraphResNet_61873298866848
// MI455X (gfx1250) — compile-verified
//
#include <hip/hip_runtime.h>

// ---------------------------------------------------------------------------
// HypergraphConv ResNet for MI455X (gfx1250, wave32).
//  - degrees once, inverted once
//  - dense X@W on the fp32 WMMA pipe (v_wmma_f32_16x16x4_f32), exact tiling
//  - incidence scatters as float4 gathers + native no-return
//    global_atomic_add_f32 (inline CDNA5 asm; L2-resident tables)
// ---------------------------------------------------------------------------

static constexpr int   kN        = 200000;
static constexpr int   kEH       = 100000;
static constexpr int   kNNZ      = 3200000;
static constexpr float kNegSlope = 0.01f;

typedef float v2f __attribute__((ext_vector_type(2)));
typedef float v8f __attribute__((ext_vector_type(8)));

// Guaranteed native non-returning fp32 global atomic add (no CAS loop, no
// return-data wait; STOREcnt-tracked, implicitly waited at S_ENDPGM).
__device__ __forceinline__ void gatomic_add_f32(float* p, float v) {
    asm volatile("global_atomic_add_f32 %0, %1, off"
                 :
                 : "v"(p), "v"(v)
                 : "memory");
}

// ---------------------------- degree kernels -------------------------------

__global__ void deg_kernel(const int* __restrict__ node, const int* __restrict__ hedge,
                           int* __restrict__ Dc, int* __restrict__ Bc, int nnz) {
    int e = blockIdx.x * blockDim.x + threadIdx.x;
    if (e >= nnz) return;
    atomicAdd(&Dc[node[e]], 1);
    atomicAdd(&Bc[hedge[e]], 1);
}

__global__ void invert_kernel(const int* __restrict__ cnt, float* __restrict__ inv, int n) {
    int i = blockIdx.x * blockDim.x + threadIdx.x;
    if (i >= n) return;
    int c = cnt[i];
    inv[i] = (c > 0) ? (1.0f / (float)c) : 0.0f;
}

// ------------------------------ layer 1 ------------------------------------
// fin=1: X@W1 is an outer product x[i] * W1[f], fout=32.

__global__ void l1_expand(const float* __restrict__ x, const float* __restrict__ W1,
                          float* __restrict__ out, int n) {
    int i = blockIdx.x * blockDim.x + threadIdx.x;   // over n*32
    if (i >= n * 32) return;
    out[i] = x[i >> 5] * W1[i & 31];
}

// --------------------------- WMMA fp32 GEMM --------------------------------
// One wave computes a 16-row tile of O[N,FOUT] = H[N,FIN] @ W[FIN,FOUT],
// accumulating with V_WMMA_F32_16X16X4_F32 (full fp32 precision).
//
// A (16x4 f32):  lanes 0-15 hold rows M=0..15; VGPR0 = K (0|2), VGPR1 = K+1.
// B (4x16 f32):  VGPR0 = row K (0|2) cols 0..15 per half-wave, VGPR1 = K+1.
// C/D (16x16):   VGPR v: lanes 0-15 -> M=v, lanes 16-31 -> M=v+8.

template <int FIN, int FOUT>
__global__ void wmma_gemm(const float* __restrict__ H, const float* __restrict__ W,
                          float* __restrict__ O) {
    const int wave = threadIdx.x >> 5;
    const int lane = threadIdx.x & 31;
    const int tile = blockIdx.x * (blockDim.x >> 5) + wave;   // 16-row tile id
    const int m    = lane & 15;
    const int kh   = lane >> 4;                                // 0 or 1
    const float* hrow = H + (size_t)(tile * 16 + m) * FIN;

#pragma unroll
    for (int ct = 0; ct < FOUT / 16; ++ct) {
        const int col = ct * 16 + m;
        v8f acc = {0.f, 0.f, 0.f, 0.f, 0.f, 0.f, 0.f, 0.f};
#pragma unroll
        for (int k0 = 0; k0 < FIN; k0 += 4) {
            const int k = k0 + 2 * kh;
            v2f a; a.x = hrow[k];                 a.y = hrow[k + 1];
            v2f b; b.x = W[k * FOUT + col];       b.y = W[(k + 1) * FOUT + col];
            acc = __builtin_amdgcn_wmma_f32_16x16x4_f32(
                false, a, false, b, (short)0, acc, false, false);
        }
        float* obase = O + (size_t)(tile * 16 + kh * 8) * FOUT + col;
#pragma unroll
        for (int v = 0; v < 8; ++v)
            obase[(size_t)v * FOUT] = acc[v];
    }
}

// ------------------------------ layer 4 ------------------------------------
// fout=1: per-node dot of 32 features with W4 (float4 vectorized).

__global__ void dot32(const float* __restrict__ h, const float* __restrict__ W4,
                      float* __restrict__ out, int n) {
    int i = blockIdx.x * blockDim.x + threadIdx.x;
    if (i >= n) return;
    const float4* r = (const float4*)(h + (size_t)i * 32);
    const float4* w = (const float4*)W4;
    float s = 0.f;
#pragma unroll
    for (int j = 0; j < 8; ++j) {
        float4 a = r[j], b = w[j];
        s += a.x * b.x + a.y * b.y + a.z * b.z + a.w * b.w;
    }
    out[i] = s;
}

// --------------------------- incidence scatters ----------------------------
// One thread per (entry, 4-float group). F = 4 << LOGF4 features per row.

template <int LOGF4>
__global__ void scatter_add_vec(const float* __restrict__ src, float* __restrict__ dst,
                                const int* __restrict__ sidx, const int* __restrict__ didx,
                                int total) {
    int tid = blockIdx.x * blockDim.x + threadIdx.x;
    if (tid >= total) return;
    const int e  = tid >> LOGF4;
    const int f4 = tid & ((1 << LOGF4) - 1);
    const int F  = 4 << LOGF4;
    const int si = sidx[e];
    const int di = didx[e];
    float4 v = *(const float4*)(src + (size_t)si * F + f4 * 4);
    float* d = dst + (size_t)di * F + f4 * 4;
    gatomic_add_f32(d + 0, v.x);
    gatomic_add_f32(d + 1, v.y);
    gatomic_add_f32(d + 2, v.z);
    gatomic_add_f32(d + 3, v.w);
}

// Same, with per-source-row scale folded in (Binv on hedge->node pass).
template <int LOGF4>
__global__ void scatter_add_vec_scaled(const float* __restrict__ src, float* __restrict__ dst,
                                       const int* __restrict__ sidx, const int* __restrict__ didx,
                                       const float* __restrict__ scale, int total) {
    int tid = blockIdx.x * blockDim.x + threadIdx.x;
    if (tid >= total) return;
    const int e  = tid >> LOGF4;
    const int f4 = tid & ((1 << LOGF4) - 1);
    const int F  = 4 << LOGF4;
    const int si = sidx[e];
    const int di = didx[e];
    const float s = scale[si];
    float4 v = *(const float4*)(src + (size_t)si * F + f4 * 4);
    float* d = dst + (size_t)di * F + f4 * 4;
    gatomic_add_f32(d + 0, s * v.x);
    gatomic_add_f32(d + 1, s * v.y);
    gatomic_add_f32(d + 2, s * v.z);
    gatomic_add_f32(d + 3, s * v.w);
}

// fout == 1 variants (layer 4).
__global__ void scatter_add_s1(const float* __restrict__ src, float* __restrict__ dst,
                               const int* __restrict__ sidx, const int* __restrict__ didx,
                               int nnz) {
    int e = blockIdx.x * blockDim.x + threadIdx.x;
    if (e >= nnz) return;
    gatomic_add_f32(&dst[didx[e]], src[sidx[e]]);
}

__global__ void scatter_add_s1_scaled(const float* __restrict__ src, float* __restrict__ dst,
                                      const int* __restrict__ sidx, const int* __restrict__ didx,
                                      const float* __restrict__ scale, int nnz) {
    int e = blockIdx.x * blockDim.x + threadIdx.x;
    if (e >= nnz) return;
    int si = sidx[e];
    gatomic_add_f32(&dst[didx[e]], src[si] * scale[si]);
}

// ------------------------------ epilogues ----------------------------------

template <int F>
__global__ void finalize_lrelu(float* __restrict__ acc, const float* __restrict__ Dinv,
                               const float* __restrict__ bias, int total) {
    int i = blockIdx.x * blockDim.x + threadIdx.x;   // over N*F
    if (i >= total) return;
    float v = acc[i] * Dinv[i / F] + bias[i % F];
    acc[i] = (v > 0.0f) ? v : kNegSlope * v;
}

__global__ void final_residual(const float* __restrict__ x, const float* __restrict__ acc,
                               const float* __restrict__ Dinv, const float* __restrict__ b4,
                               float* __restrict__ out, int n) {
    int i = blockIdx.x * blockDim.x + threadIdx.x;
    if (i >= n) return;
    out[i] = x[i] + (acc[i] * Dinv[i] + b4[0]);
}

// ------------------------------- launcher ----------------------------------

static inline int cdiv(int a, int b) { return (a + b - 1) / b; }

extern "C" void kernel_launch(void* const* d_in, const int* in_sizes, int n_in,
                              void* d_out, int out_size, void* d_ws, size_t ws_size,
                              hipStream_t stream) {
    (void)in_sizes; (void)n_in; (void)out_size; (void)ws_size;

    const float* x     = (const float*)d_in[0];
    const int*   ei    = (const int*)d_in[1];
    const int*   node  = ei;            // edge_index[0]
    const int*   hedge = ei + kNNZ;     // edge_index[1]
    const float* W1 = (const float*)d_in[2];
    const float* b1 = (const float*)d_in[3];
    const float* W2 = (const float*)d_in[4];
    const float* b2 = (const float*)d_in[5];
    const float* W3 = (const float*)d_in[6];
    const float* b3 = (const float*)d_in[7];
    const float* W4 = (const float*)d_in[8];
    const float* b4 = (const float*)d_in[9];
    float*       out = (float*)d_out;

    // workspace layout (floats): A[N*64] | B[N*64] | xe[EH*64] | Dinv[N] | Binv[EH] | Dcnt[N] | Bcnt[EH]
    float* bufA = (float*)d_ws;
    float* bufB = bufA + (size_t)kN * 64;
    float* xe   = bufB + (size_t)kN * 64;
    float* Dinv = xe   + (size_t)kEH * 64;
    float* Binv = Dinv + kN;
    int*   Dcnt = (int*)(Binv + kEH);
    int*   Bcnt = Dcnt + kN;

    const int TB = 256;

    // --- degrees (once, reused by all 4 layers) ---
    hipMemsetAsync(Dcnt, 0, sizeof(int) * (size_t)kN, stream);
    hipMemsetAsync(Bcnt, 0, sizeof(int) * (size_t)kEH, stream);
    deg_kernel<<<cdiv(kNNZ, TB), TB, 0, stream>>>(node, hedge, Dcnt, Bcnt, kNNZ);
    invert_kernel<<<cdiv(kN, TB), TB, 0, stream>>>(Dcnt, Dinv, kN);
    invert_kernel<<<cdiv(kEH, TB), TB, 0, stream>>>(Bcnt, Binv, kEH);

    // --- layer 1: 1 -> 32 ---
    l1_expand<<<cdiv(kN * 32, TB), TB, 0, stream>>>(x, W1, bufB, kN);
    hipMemsetAsync(xe,   0, sizeof(float) * (size_t)kEH * 32, stream);
    hipMemsetAsync(bufA, 0, sizeof(float) * (size_t)kN * 32, stream);
    scatter_add_vec<3><<<cdiv(kNNZ * 8, TB), TB, 0, stream>>>(bufB, xe, node, hedge, kNNZ * 8);
    scatter_add_vec_scaled<3><<<cdiv(kNNZ * 8, TB), TB, 0, stream>>>(xe, bufA, hedge, node, Binv, kNNZ * 8);
    finalize_lrelu<32><<<cdiv(kN * 32, TB), TB, 0, stream>>>(bufA, Dinv, b1, kN * 32);

    // --- layer 2: 32 -> 64 (WMMA fp32) ---
    wmma_gemm<32, 64><<<(kN / 16) / 4, 128, 0, stream>>>(bufA, W2, bufB);
    hipMemsetAsync(xe,   0, sizeof(float) * (size_t)kEH * 64, stream);
    hipMemsetAsync(bufA, 0, sizeof(float) * (size_t)kN * 64, stream);
    scatter_add_vec<4><<<cdiv(kNNZ * 16, TB), TB, 0, stream>>>(bufB, xe, node, hedge, kNNZ * 16);
    scatter_add_vec_scaled<4><<<cdiv(kNNZ * 16, TB), TB, 0, stream>>>(xe, bufA, hedge, node, Binv, kNNZ * 16);
    finalize_lrelu<64><<<cdiv(kN * 64, TB), TB, 0, stream>>>(bufA, Dinv, b2, kN * 64);

    // --- layer 3: 64 -> 32 (WMMA fp32) ---
    wmma_gemm<64, 32><<<(kN / 16) / 4, 128, 0, stream>>>(bufA, W3, bufB);
    hipMemsetAsync(xe,   0, sizeof(float) * (size_t)kEH * 32, stream);
    hipMemsetAsync(bufA, 0, sizeof(float) * (size_t)kN * 32, stream);
    scatter_add_vec<3><<<cdiv(kNNZ * 8, TB), TB, 0, stream>>>(bufB, xe, node, hedge, kNNZ * 8);
    scatter_add_vec_scaled<3><<<cdiv(kNNZ * 8, TB), TB, 0, stream>>>(xe, bufA, hedge, node, Binv, kNNZ * 8);
    finalize_lrelu<32><<<cdiv(kN * 32, TB), TB, 0, stream>>>(bufA, Dinv, b3, kN * 32);

    // --- layer 4: 32 -> 1 + residual ---
    dot32<<<cdiv(kN, TB), TB, 0, stream>>>(bufA, W4, bufB, kN);
    hipMemsetAsync(xe,   0, sizeof(float) * (size_t)kEH, stream);
    hipMemsetAsync(bufA, 0, sizeof(float) * (size_t)kN, stream);
    scatter_add_s1<<<cdiv(kNNZ, TB), TB, 0, stream>>>(bufB, xe, node, hedge, kNNZ);
    scatter_add_s1_scaled<<<cdiv(kNNZ, TB), TB, 0, stream>>>(xe, bufA, hedge, node, Binv, kNNZ);
    final_residual<<<cdiv(kN, TB), TB, 0, stream>>>(x, bufA, Dinv, b4, out, kN);
}